// RNNTLoss_4956392259585
// MI455X (gfx1250) — compile-verified
//
#include <hip/hip_runtime.h>

#define Bb 4
#define Tt 512
#define Uu 65
#define Vv 1024

typedef float v2f __attribute__((ext_vector_type(2)));
typedef float v8f __attribute__((ext_vector_type(8)));

__device__ __forceinline__ float neg_inf() { return -__builtin_inff(); }

// Numerically stable log(exp(a)+exp(b)); identity element is -inf.
__device__ __forceinline__ float laddexp(float a, float b) {
    float m = fmaxf(a, b);
    float d = fminf(a, b) - m;                 // <= 0 (NaN only if both -inf)
    float r = m + __logf(1.0f + __expf(d));
    return (m == neg_inf()) ? m : r;
}

// ---------------------------------------------------------------------------
// Pass 1: per-row logsumexp over V; emit blank/emit log-probs to workspace.
// One wave (32 lanes) per row of 1024 floats. Lane j of load i holds
// elements [i*128 + 4*lane .. +3] (b128 loads, fully coalesced, 512B/instr).
// ---------------------------------------------------------------------------
__global__ __launch_bounds__(256) void rnnt_pass1(
        const float* __restrict__ logits, const int* __restrict__ targets,
        float* __restrict__ blankW, float* __restrict__ emitW) {
    const int lane = threadIdx.x & 31;
    const int wave = threadIdx.x >> 5;
    const int r    = blockIdx.x * 8 + wave;         // row id, exact grid
    const int TU   = Tt * Uu;
    const int b    = r / TU;
    const int rem  = r - b * TU;
    const int t    = rem / Uu;
    const int u    = rem - t * Uu;

    const float4* __restrict__ rp =
        (const float4*)(logits + (size_t)r * Vv);

    float4 x[8];
#pragma unroll
    for (int j = 0; j < 8; ++j) x[j] = rp[j * 32 + lane];

    // ---- wave max (exact, 5 shuffle steps) ----
    float m = neg_inf();
#pragma unroll
    for (int j = 0; j < 8; ++j) {
        m = fmaxf(m, fmaxf(fmaxf(x[j].x, x[j].y), fmaxf(x[j].z, x[j].w)));
    }
#pragma unroll
    for (int d = 16; d >= 1; d >>= 1)
        m = fmaxf(m, __shfl_xor(m, d, 32));

    // ---- per-lane partial sum of exp(x - m) ----
    float s = 0.0f;
#pragma unroll
    for (int j = 0; j < 8; ++j) {
        s += __expf(x[j].x - m);
        s += __expf(x[j].y - m);
        s += __expf(x[j].z - m);
        s += __expf(x[j].w - m);
    }

    // ---- cross-lane sum via V_WMMA_F32_16X16X4_F32 ----
    // A (16x4): lane l<16 -> A[l][0]=s_l, A[l][1]=0 ; lane l>=16 -> A[l-16][2]=s_l.
    // B = ones(4x16)  =>  C[m][n] = s_m + s_{m+16} for every n.
    v2f Am; Am[0] = s;    Am[1] = 0.0f;
    v2f Bm; Bm[0] = 1.0f; Bm[1] = 1.0f;
    v8f Cm = {};
    Cm = __builtin_amdgcn_wmma_f32_16x16x4_f32(
        /*neg_a=*/false, Am, /*neg_b=*/false, Bm,
        /*c_mod=*/(short)0, Cm, /*reuse_a=*/false, /*reuse_b=*/false);
    // C VGPR r: lanes 0-15 hold t_r = s_r + s_{r+16}; lanes 16-31 hold t_{r+8}.
    float half = Cm[0] + Cm[1] + Cm[2] + Cm[3] + Cm[4] + Cm[5] + Cm[6] + Cm[7];
    float total = half + __shfl_xor(half, 16, 32);   // full 32-lane sum

    const float lse = m + __logf(total);

    if (lane == 0) {
        const float x0 = logits[(size_t)r * Vv];           // BLANK = index 0
        blankW[r] = x0 - lse;
        if (u < Uu - 1) {
            const int tgt = targets[b * (Uu - 1) + u];     // in [1, V)
            const float xt = logits[(size_t)r * Vv + tgt]; // L2 hit (just streamed)
            emitW[(b * Tt + t) * (Uu - 1) + u] = xt - lse;
        }
    }
}

// ---------------------------------------------------------------------------
// Pass 2: alpha scan.  One wave per batch element; lane l owns u=2l, 2l+1;
// u=64 handled as a scalar epilogue.  Per t-step: cs = exclusive +scan of
// emit, then a = cs + inclusive logaddexp-scan of (alpha_prev+blank - cs).
// ---------------------------------------------------------------------------
__global__ __launch_bounds__(128) void rnnt_pass2(
        const float* __restrict__ blankW, const float* __restrict__ emitW,
        const int* __restrict__ input_lengths,
        const int* __restrict__ target_lengths, float* __restrict__ out) {
    __shared__ float finals[Bb];
    const int lane = threadIdx.x & 31;
    const int b    = threadIdx.x >> 5;          // wave id == batch id

    const float* __restrict__ blk = blankW + (size_t)b * Tt * Uu;
    const float* __restrict__ emt = emitW  + (size_t)b * Tt * (Uu - 1);
    const int tl = input_lengths[b] - 1;        // final time index
    const int ul = target_lengths[b];           // final u index

    float a_lo = 0.0f, a_hi = 0.0f, a64 = 0.0f; // alpha[2l], alpha[2l+1], alpha[64]

    for (int t = 0; t < Tt; ++t) {
        if (t == 0) {
            // alpha0[u] = sum_{j<u} emit[0,j]  (exclusive cumsum)
            const float e0 = emt[2 * lane], e1 = emt[2 * lane + 1];
            float v = e0 + e1;                  // lane aggregate
#pragma unroll
            for (int d = 1; d < 32; d <<= 1) {
                float w = __shfl_up(v, d, 32);
                v += (lane >= d) ? w : 0.0f;
            }
            float excl = __shfl_up(v, 1, 32);
            excl = (lane == 0) ? 0.0f : excl;   // sum over previous lanes
            a_lo = excl;                        // inc[2l-1] = excl
            a_hi = excl + e0;                   // inc[2l]
            a64  = __shfl(excl + e0 + e1, 31, 32);   // inc[63]
        } else {
            // loads for this step
            const float bp0  = blk[(t - 1) * Uu + 2 * lane];
            const float bp1  = blk[(t - 1) * Uu + 2 * lane + 1];
            const float bp64 = blk[(t - 1) * Uu + 64];
            const float et0  = emt[t * (Uu - 1) + 2 * lane];
            const float et1  = emt[t * (Uu - 1) + 2 * lane + 1];

            // b_vec = alpha_prev + blank_prev
            const float b0  = a_lo + bp0;
            const float b1  = a_hi + bp1;
            const float b64 = a64 + bp64;

            // cs = exclusive cumsum of emit_t
            float v = et0 + et1;
#pragma unroll
            for (int d = 1; d < 32; d <<= 1) {
                float w = __shfl_up(v, d, 32);
                v += (lane >= d) ? w : 0.0f;
            }
            float csx = __shfl_up(v, 1, 32);
            csx = (lane == 0) ? 0.0f : csx;
            const float cs0 = csx;              // cs[2l]
            const float cs1 = csx + et0;        // cs[2l+1]

            // z = b - cs ; inclusive logaddexp scan over u=0..63
            const float z0 = b0 - cs0;
            const float z1 = b1 - cs1;
            float p = laddexp(z0, z1);          // lane aggregate
#pragma unroll
            for (int d = 1; d < 32; d <<= 1) {
                float w = __shfl_up(p, d, 32);
                float c = laddexp(p, w);
                p = (lane >= d) ? c : p;
            }
            float lex = __shfl_up(p, 1, 32);
            lex = (lane == 0) ? neg_inf() : lex;
            const float l0 = laddexp(lex, z0);
            const float l1 = laddexp(l0, z1);
            a_lo = cs0 + l0;
            a_hi = cs1 + l1;

            // epilogue u = 64: a[64] = logaddexp(b64, a[63] + emit[63])
            const float a63 = __shfl(a_hi, 31, 32);
            const float e63 = __shfl(et1, 31, 32);
            a64 = laddexp(b64, a63 + e63);
        }

        if (t == tl) {
            float aul = (ul == 64)
                ? a64
                : __shfl((ul & 1) ? a_hi : a_lo, ul >> 1, 32);
            if (lane == 0) finals[b] = aul + blk[t * Uu + ul];
        }
    }

    __syncthreads();
    if (threadIdx.x == 0) {
        float acc = 0.0f;
#pragma unroll
        for (int i = 0; i < Bb; ++i) acc += finals[i];
        out[0] = -acc / (float)Bb;              // mean(-final), deterministic
    }
}

// ---------------------------------------------------------------------------
extern "C" void kernel_launch(void* const* d_in, const int* in_sizes, int n_in,
                              void* d_out, int out_size, void* d_ws, size_t ws_size,
                              hipStream_t stream) {
    const float* logits   = (const float*)d_in[0];
    const int*   targets  = (const int*)d_in[1];
    const int*   inlen    = (const int*)d_in[2];
    const int*   tlen     = (const int*)d_in[3];
    float* out = (float*)d_out;

    float* blankW = (float*)d_ws;                       // B*T*U floats
    float* emitW  = blankW + (size_t)Bb * Tt * Uu;      // B*T*(U-1) floats

    const int rows = Bb * Tt * Uu;                      // 133,120 (divisible by 8)
    rnnt_pass1<<<rows / 8, 256, 0, stream>>>(logits, targets, blankW, emitW);
    rnnt_pass2<<<1, 128, 0, stream>>>(blankW, emitW, inlen, tlen, out);
}